// QuantumQuadLayer_68822555951765
// MI455X (gfx1250) — compile-verified
//
#include <hip/hip_runtime.h>
#include <hip/hip_bf16.h>

typedef float v2f __attribute__((ext_vector_type(2)));
typedef float v8f __attribute__((ext_vector_type(8)));

#define NQ 4
#define DIM 16
#define NL 2
#define WPB 8   // waves per block (wave32)

// ---------------------------------------------------------------------------
// Setup kernel (1 wave, runs once): build the fixed real-symmetric 16x16
// matrix G such that out_m = t_m^T G t_m, where t is the real tensor-product
// vector of the RX embedding and G = phase-adjusted V^H Z0 V for the fixed
// StronglyEntanglingLayers unitary V.
// ---------------------------------------------------------------------------
__global__ __launch_bounds__(32) void quantum_setup_kernel(
    const float* __restrict__ wts,   // (2, 4, 3) weights
    float* __restrict__ Gout)        // 16x16 floats
{
    __shared__ float Vre[DIM][DIM];
    __shared__ float Vim[DIM][DIM];

    const int tid = threadIdx.x;

    if (tid < DIM) {
        float ar[DIM], ai[DIM];
        #pragma unroll
        for (int i = 0; i < DIM; ++i) { ar[i] = (i == tid) ? 1.f : 0.f; ai[i] = 0.f; }

        #pragma unroll
        for (int l = 0; l < NL; ++l) {
            // Rot(phi, theta, omega) on each wire
            #pragma unroll
            for (int w = 0; w < NQ; ++w) {
                float phi = wts[(l * NQ + w) * 3 + 0];
                float th  = wts[(l * NQ + w) * 3 + 1];
                float om  = wts[(l * NQ + w) * 3 + 2];
                float ct, st, ca, sa, cb, sb;
                __sincosf(0.5f * th, &st, &ct);
                __sincosf(0.5f * (phi + om), &sa, &ca);   // ep = exp(-i a) = (ca, -sa)
                __sincosf(0.5f * (phi - om), &sb, &cb);   // em = exp(+i b) = (cb,  sb)
                float u00r =  ca * ct, u00i = -sa * ct;   // ep * c
                float u01r = -cb * st, u01i = -sb * st;   // -em * s
                float u10r =  cb * st, u10i = -sb * st;   // conj(em) * s
                float u11r =  ca * ct, u11i =  sa * ct;   // conj(ep) * c
                const int mask = 1 << (3 - w);
                #pragma unroll
                for (int j = 0; j < DIM; ++j) {
                    if ((j & mask) == 0) {
                        const int j1 = j | mask;
                        float x0r = ar[j], x0i = ai[j], x1r = ar[j1], x1i = ai[j1];
                        ar[j]  = u00r * x0r - u00i * x0i + u01r * x1r - u01i * x1i;
                        ai[j]  = u00r * x0i + u00i * x0r + u01r * x1i + u01i * x1r;
                        ar[j1] = u10r * x0r - u10i * x0i + u11r * x1r - u11i * x1i;
                        ai[j1] = u10r * x0i + u10i * x0r + u11r * x1i + u11i * x1r;
                    }
                }
            }
            // CNOT ring, range r = l % 3 + 1
            const int rng = (l % (NQ - 1)) + 1;
            #pragma unroll
            for (int w = 0; w < NQ; ++w) {
                const int tq = (w + rng) % NQ;
                const int mc = 1 << (3 - w);
                const int mt = 1 << (3 - tq);
                #pragma unroll
                for (int j = 0; j < DIM; ++j) {
                    if ((j & mc) != 0 && (j & mt) == 0) {
                        const int j1 = j | mt;
                        float tr = ar[j]; ar[j] = ar[j1]; ar[j1] = tr;
                        float ti = ai[j]; ai[j] = ai[j1]; ai[j1] = ti;
                    }
                }
            }
        }
        #pragma unroll
        for (int i = 0; i < DIM; ++i) { Vre[i][tid] = ar[i]; Vim[i][tid] = ai[i]; }
    }
    __syncthreads();

    // G[j][k] = Re[ i^(p(j)-p(k)) * H_jk ],  H = V^H Z0 V  (real symmetric)
    if (tid < DIM) {
        const int pj = __popc(tid);
        for (int k = 0; k < DIM; ++k) {
            float hr = 0.f, hm = 0.f;
            #pragma unroll
            for (int i = 0; i < DIM; ++i) {
                float z = ((i >> 3) & 1) ? -1.f : 1.f;
                float a_r = Vre[i][tid], a_i = Vim[i][tid];
                float b_r = Vre[i][k],   b_i = Vim[i][k];
                hr += z * (a_r * b_r + a_i * b_i);     // Re(conj(a)*b)
                hm += z * (a_r * b_i - a_i * b_r);     // Im(conj(a)*b)
            }
            const int d = (pj - __popc(k)) & 3;
            Gout[tid * DIM + k] = (d == 0) ? hr : (d == 1) ? -hm : (d == 2) ? -hr : hm;
        }
    }
}

// ---------------------------------------------------------------------------
// Main kernel: one 16-sample tile per wave.
//   t_j = prod_w (bit_w(j) ? sin(x_w/2) : cos(x_w/2))
//   W = T(16x16) * G(16x16) via 4 chained V_WMMA_F32_16X16X4_F32
//   out_m = <t_m, W[m,:]>
// ---------------------------------------------------------------------------
__global__ __launch_bounds__(256) void quantum_quad_kernel(
    const float* __restrict__ x,     // (M, 4) angles
    const float* __restrict__ Gmat,  // 16x16 from setup kernel
    float* __restrict__ out, int M)
{
    __shared__ float Gs[DIM][DIM];
    __shared__ float Wt[WPB][DIM][DIM + 1];

    const int tid = threadIdx.x;

    // Cooperative load of G into LDS (one float per thread, L2 broadcast).
    Gs[tid >> 4][tid & 15] = Gmat[tid];
    __syncthreads();

    const int lane = tid & 31;
    const int wv   = tid >> 5;
    const long long tile = (long long)blockIdx.x * WPB + wv;
    const long long m0   = tile * DIM;
    const int  srow   = lane & 15;
    const bool hiHalf = lane >= 16;
    const long long m  = m0 + srow;
    const long long mcl = (m < (long long)M) ? m : (long long)(M - 1);

    float4 ang = ((const float4*)x)[mcl];
    float c0, s0, c1, s1, c2, s2, c3, s3;
    __sincosf(0.5f * ang.x, &s0, &c0);
    __sincosf(0.5f * ang.y, &s1, &c1);
    __sincosf(0.5f * ang.z, &s2, &c2);
    __sincosf(0.5f * ang.w, &s3, &c3);

    // t_j = prod_w (bit_w(j) ? s_w : c_w),  j = i0*8 + i1*4 + i2*2 + i3
    float p01[4] = {c0 * c1, c0 * s1, s0 * c1, s0 * s1};
    float p23[4] = {c2 * c3, c2 * s3, s2 * c3, s2 * s3};
    float t[16];
    #pragma unroll
    for (int j = 0; j < 16; ++j) t[j] = p01[j >> 2] * p23[j & 3];

    const int n = lane & 15;
    v8f acc = {0.f, 0.f, 0.f, 0.f, 0.f, 0.f, 0.f, 0.f};

    // A layout (32-bit 16x4): lanes 0-15 hold K=0,1 in v0,v1; lanes 16-31 K=2,3.
    // B layout (32-bit 4x16): lanes 0-15 hold K=0,1; lanes 16-31 K=2,3; N = lane%16.
    #pragma unroll
    for (int s = 0; s < 4; ++s) {
        v2f a, b;
        a.x = hiHalf ? t[4 * s + 2] : t[4 * s + 0];
        a.y = hiHalf ? t[4 * s + 3] : t[4 * s + 1];
        const int k0 = 4 * s + (hiHalf ? 2 : 0);
        b.x = Gs[k0][n];
        b.y = Gs[k0 + 1][n];
        acc = __builtin_amdgcn_wmma_f32_16x16x4_f32(
            /*neg_a=*/false, a, /*neg_b=*/false, b,
            /*c_mod=*/(short)0, acc, /*reuse_a=*/false, /*reuse_b=*/false);
    }

    // Spill D tile: vgpr r, lanes 0-15 -> (M=r, N=lane); lanes 16-31 -> (M=r+8, N=lane-16)
    const int rowBase = hiHalf ? 8 : 0;
    #pragma unroll
    for (int r = 0; r < 8; ++r) Wt[wv][rowBase + r][n] = acc[r];
    __syncthreads();

    // out_m = sum_j t[j] * W[m][j]
    if (lane < 16 && m < (long long)M) {
        float o = 0.f;
        #pragma unroll
        for (int j = 0; j < 16; ++j) o += t[j] * Wt[wv][srow][j];
        out[m] = o;
    }
}

extern "C" void kernel_launch(void* const* d_in, const int* in_sizes, int n_in,
                              void* d_out, int out_size, void* d_ws, size_t ws_size,
                              hipStream_t stream) {
    const float* x   = (const float*)d_in[0];   // (64, 8192, 4) f32
    const float* wts = (const float*)d_in[1];   // (2, 4, 3) f32
    float* out  = (float*)d_out;                // (64, 8192, 1) f32
    float* Gmat = (float*)d_ws;                 // 16x16 scratch
    const int M = out_size;                     // 524288 samples

    quantum_setup_kernel<<<1, 32, 0, stream>>>(wts, Gmat);

    const int tiles  = (M + DIM - 1) / DIM;
    const int blocks = (tiles + WPB - 1) / WPB;
    quantum_quad_kernel<<<blocks, 256, 0, stream>>>(x, Gmat, out, M);
}